// AutoLayer_53060025975244
// MI455X (gfx1250) — compile-verified
//
#include <hip/hip_runtime.h>
#include <hip/hip_bf16.h>

#define N_NODESC 100000
#define N_EDGESC 1600000
#define DDIM     128

constexpr float ALPHA_  = 0.1f;
constexpr float BETA_   = 0.5f;
constexpr float LN_EPS_ = 1e-5f;

typedef float v2f __attribute__((ext_vector_type(2)));
typedef float v8f __attribute__((ext_vector_type(8)));

// ---------------------------------------------------------------------------
// Kernel 0: zero the hidden accumulator (d_ws), float4-vectorized.
// ---------------------------------------------------------------------------
__global__ void zero_f4_kernel(float4* __restrict__ p, int n4) {
    int i = blockIdx.x * blockDim.x + threadIdx.x;
    if (i < n4) p[i] = make_float4(0.f, 0.f, 0.f, 0.f);
}

// ---------------------------------------------------------------------------
// Kernel 1: COO SPMM scatter. One wave32 per edge; lane l covers columns
// l, l+32, l+64, l+96 -> 128B coalesced x-row loads and coalesced
// global_atomic_add_f32 into hidden[dst]. x and hidden are both L2-resident
// (51.2 MB each vs 192 MB L2), so this is L2-bandwidth bound, not HBM bound.
// ---------------------------------------------------------------------------
__global__ void spmm_scatter_kernel(const int* __restrict__ ei,
                                    const float* __restrict__ ew,
                                    const float* __restrict__ x,
                                    float* __restrict__ hidden) {
    unsigned gid  = blockIdx.x * blockDim.x + threadIdx.x;
    unsigned e    = gid >> 5;
    unsigned lane = gid & 31u;
    if (e >= N_EDGESC) return;
    int   dst = ei[e];             // edge_index[0] = dst
    int   src = ei[N_EDGESC + e];  // edge_index[1] = src
    float w   = ew[e];
    const float* xs = x + (size_t)src * DDIM;
    float*       h  = hidden + (size_t)dst * DDIM;
#pragma unroll
    for (int i = 0; i < 4; ++i) {
        unsigned c = lane + 32u * i;
        atomicAdd(&h[c], w * xs[c]);
    }
}

// ---------------------------------------------------------------------------
// Kernel 2: fused combine + beta-GEMM + LayerNorm + ReLU + residual GEMM.
// Block = 256 threads (8 wave32s) handles 16 node rows (6250 blocks exactly).
// Wave w computes the 16x16 output tile for columns [16w, 16w+16) of BOTH
// GEMMs with V_WMMA_F32_16X16X4_F32, K-loop over DDIM=128 in steps of 4.
// ---------------------------------------------------------------------------
__global__ void __launch_bounds__(256)
fused_tail_kernel(const float* __restrict__ hidden,
                  const float* __restrict__ init_x,
                  const float* __restrict__ x,
                  const float* __restrict__ Wlin, const float* __restrict__ blin,
                  const float* __restrict__ Wres, const float* __restrict__ bres,
                  const float* __restrict__ gamma, const float* __restrict__ beta,
                  float* __restrict__ out) {
    __shared__ float sA[16][132];   // alpha-combined hidden tile (pitch 132: conflict-free frags)
    __shared__ float sX[16][132];   // layer-input tile for the residual GEMM
    __shared__ float sO[16][132];   // pre-LayerNorm output tile
    __shared__ float sMu[16];
    __shared__ float sRstd[16];

    const int    tid     = threadIdx.x;
    const size_t rowBase = (size_t)blockIdx.x * 16;

    // Cooperative, coalesced tile load with the GCNII identity-mapping combine.
    for (int idx = tid; idx < 16 * DDIM; idx += 256) {
        int r = idx >> 7, c = idx & 127;
        size_t g = (rowBase + r) * DDIM + c;
        sA[r][c] = (1.0f - ALPHA_) * hidden[g] + ALPHA_ * init_x[g];
        sX[r][c] = x[g];
    }
    __syncthreads();

    const int wave = tid >> 5;
    const int lane = tid & 31;
    const int n0   = wave * 16;     // this wave's output-column tile
    const int half = lane >> 4;     // 0: lanes 0-15, 1: lanes 16-31
    const int l15  = lane & 15;
    const int ksel = half * 2;      // A/B fragments: upper half-wave holds K+2,K+3

    // B fragment source: B[k][n] = W^T[k][n] = W[n][k]; per-lane row pointer.
    const float* WLp = Wlin + (size_t)(n0 + l15) * DDIM;
    const float* WRp = Wres + (size_t)(n0 + l15) * DDIM;
    __builtin_prefetch(WLp, 0, 1);   // global_prefetch_b8 (W stays L2/WGP$-resident)
    __builtin_prefetch(WRp, 0, 1);

    v8f cl = {};   // hidden @ Wlin^T accumulator (16x16 f32 tile)
    v8f cr = {};   // x      @ Wres^T accumulator

#pragma unroll 4
    for (int k0 = 0; k0 < DDIM; k0 += 4) {
        const int k = k0 + ksel;
        v2f aL, aX, bL, bR;
        aL.x = sA[l15][k];  aL.y = sA[l15][k + 1];
        aX.x = sX[l15][k];  aX.y = sX[l15][k + 1];
        bL.x = WLp[k];      bL.y = WLp[k + 1];
        bR.x = WRp[k];      bR.y = WRp[k + 1];
        cl = __builtin_amdgcn_wmma_f32_16x16x4_f32(false, aL, false, bL,
                                                   (short)0, cl, false, false);
        cr = __builtin_amdgcn_wmma_f32_16x16x4_f32(false, aX, false, bR,
                                                   (short)0, cr, false, false);
    }

    // C/D layout: VGPR j -> row M = j + 8*half, col N = n0 + l15.
    const int   ncol  = n0 + l15;
    const int   mbase = half * 8;
    const float bl    = blin[ncol];
#pragma unroll
    for (int j = 0; j < 8; ++j) {
        int m = mbase + j;
        sO[m][ncol] = BETA_ * (cl[j] + bl) + (1.0f - BETA_) * sA[m][ncol];
    }
    __syncthreads();

    // Row-wise LayerNorm statistics (16 rows; one thread per row is plenty).
    if (tid < 16) {
        float s = 0.f, ss = 0.f;
        for (int c = 0; c < DDIM; ++c) { float v = sO[tid][c]; s += v; ss += v * v; }
        float mu  = s * (1.0f / DDIM);
        float var = ss * (1.0f / DDIM) - mu * mu;
        sMu[tid]   = mu;
        sRstd[tid] = rsqrtf(var + LN_EPS_);
    }
    __syncthreads();

    // LN + ReLU + residual-projection add, straight from the live cr accumulators.
    const float br = bres[ncol];
    const float g_ = gamma[ncol];
    const float bb = beta[ncol];
#pragma unroll
    for (int j = 0; j < 8; ++j) {
        int   m  = mbase + j;
        float v  = sO[m][ncol];
        float ln = g_ * (v - sMu[m]) * sRstd[m] + bb;
        ln = fmaxf(ln, 0.0f);
        out[(rowBase + m) * DDIM + ncol] = ln + cr[j] + br;
    }
}

// ---------------------------------------------------------------------------
extern "C" void kernel_launch(void* const* d_in, const int* in_sizes, int n_in,
                              void* d_out, int out_size, void* d_ws, size_t ws_size,
                              hipStream_t stream) {
    const int*   ei     = (const int*)d_in[0];    // edge_index [2, E]
    const float* ew     = (const float*)d_in[1];  // edge_weight [E]
    const float* x      = (const float*)d_in[2];  // [N, D]
    const float* init_x = (const float*)d_in[3];  // [N, D]
    const float* Wlin   = (const float*)d_in[4];  // [D, D]
    const float* blin   = (const float*)d_in[5];  // [D]
    const float* Wres   = (const float*)d_in[6];  // [D, D]
    const float* bres   = (const float*)d_in[7];  // [D]
    const float* gamma  = (const float*)d_in[8];  // [D]
    const float* beta   = (const float*)d_in[9];  // [D]
    float*       out    = (float*)d_out;
    float*       hidden = (float*)d_ws;           // N*D floats = 51.2 MB scratch

    // 0) zero the SPMM accumulator
    const int n4 = N_NODESC * DDIM / 4;
    zero_f4_kernel<<<(n4 + 255) / 256, 256, 0, stream>>>((float4*)hidden, n4);

    // 1) edge scatter: 8 edges per 256-thread block, E/8 = 200000 blocks exactly
    spmm_scatter_kernel<<<N_EDGESC / 8, 256, 0, stream>>>(ei, ew, x, hidden);

    // 2) fused dense tail: 16 rows per block, N/16 = 6250 blocks exactly
    fused_tail_kernel<<<N_NODESC / 16, 256, 0, stream>>>(
        hidden, init_x, x, Wlin, blin, Wres, bres, gamma, beta, out);
}